// DistributionFreeModel_42021960024549
// MI455X (gfx1250) — compile-verified
//
#include <hip/hip_runtime.h>
#include <stdint.h>

// ---------------- problem constants (from reference) ----------------
#define B_DIM 32768
#define F_DIM 512
#define P_DIM 118   // (9-2)*9 + 2*9*3 + 1

// ---------------- gfx1250 feature detection ----------------
#if defined(__has_builtin)
#  if __has_builtin(__builtin_amdgcn_global_load_async_to_lds_b32)
#    define HAVE_ASYNC_LDS 1
#  endif
#endif
#ifndef HAVE_ASYNC_LDS
#  define HAVE_ASYNC_LDS 0
#endif

// address-space-qualified pointee typedefs (attribute binds to the typedef'd
// type, so `as1_int*` is unambiguously "pointer to AS1 int")
typedef __attribute__((address_space(1))) int as1_int;
typedef __attribute__((address_space(3))) int as3_int;

// hardware tanh (CDNA5 tanh-insts) with exp2/rcp fallback
__device__ __forceinline__ float fast_tanh(float x) {
#if defined(__has_builtin) && __has_builtin(__builtin_amdgcn_tanhf)
  return __builtin_amdgcn_tanhf(x);
#else
  // tanh(x) = 1 - 2/(exp(2x)+1);  2*log2(e) = 2.885390081777927
  float e = __builtin_amdgcn_exp2f(2.885390081777927f * x);
  return 1.0f - 2.0f * __builtin_amdgcn_rcpf(e + 1.0f);
#endif
}

__device__ __forceinline__ float fast_sigmoid(float x) {
  // 1/(1+exp(-x));  log2(e) = 1.4426950408889634
  float e = __builtin_amdgcn_exp2f(-1.4426950408889634f * x);
  return __builtin_amdgcn_rcpf(1.0f + e);
}

// one-lane 4B async copy global -> LDS (gfx1250 GLOBAL_LOAD_ASYNC_TO_LDS_B32)
__device__ __forceinline__ void async_ld32(const float* g, float* l) {
#if HAVE_ASYNC_LDS
  __builtin_amdgcn_global_load_async_to_lds_b32(
      (as1_int*)(unsigned long long)(uintptr_t)g,   // global src (AS1)
      (as3_int*)(unsigned int)(uintptr_t)l,         // LDS dst (AS3; low 32 bits of generic ptr)
      0, 0);
#else
  *l = *g;  // synchronous fallback (still correct)
#endif
}

// ---------------- per-element MLP (params in registers) ----------------
__device__ __forceinline__ float mlp_eval(float x, const float* __restrict__ prm) {
  // layer 0: 1 -> 3, then gate
  float h0 = fmaf(prm[0], x, prm[3]);
  float h1 = fmaf(prm[1], x, prm[4]);
  float h2 = fmaf(prm[2], x, prm[5]);
  h0 = fmaf(prm[6], fast_tanh(h0), h0);
  h1 = fmaf(prm[7], fast_tanh(h1), h1);
  h2 = fmaf(prm[8], fast_tanh(h2), h2);
  // 7 middle layers: 3x3 matvec + bias, then gate
#pragma unroll
  for (int l = 0; l < 7; ++l) {
    const int p = 9 + 15 * l;
    float t0 = fmaf(prm[p + 0], h0, fmaf(prm[p + 1], h1, fmaf(prm[p + 2], h2, prm[p + 9])));
    float t1 = fmaf(prm[p + 3], h0, fmaf(prm[p + 4], h1, fmaf(prm[p + 5], h2, prm[p + 10])));
    float t2 = fmaf(prm[p + 6], h0, fmaf(prm[p + 7], h1, fmaf(prm[p + 8], h2, prm[p + 11])));
    h0 = fmaf(prm[p + 12], fast_tanh(t0), t0);
    h1 = fmaf(prm[p + 13], fast_tanh(t1), t1);
    h2 = fmaf(prm[p + 14], fast_tanh(t2), t2);
  }
  // final layer: 3 -> 1, sigmoid
  float y = fmaf(prm[114], h0, fmaf(prm[115], h1, fmaf(prm[116], h2, prm[117])));
  return fast_sigmoid(y);
}

// ---------------- main evaluation kernel (placed FIRST so the disasm
// snippet shows it: looking for global_load_async_to_lds_b32 + v_tanh_f32) ----
#define WAVES 8
#define ROWS_PER_BLOCK 256
#define PAIRS (ROWS_PER_BLOCK / WAVES / 2)  // 16 iterations, 2 rows each

__global__ __launch_bounds__(256) void cdf_main(const float* __restrict__ in,
                                                const float* __restrict__ pt,
                                                float* __restrict__ out) {
  // per-wave double-buffered input stage: [wave][buf][rowpair][lane]
  __shared__ float stage[WAVES][2][2][32];

  const int lane  = threadIdx.x & 31;
  const int wave  = threadIdx.x >> 5;
  const int fbase = blockIdx.x * 32;
  const int f     = fbase + lane;
  const int row0  = blockIdx.y * ROWS_PER_BLOCK + wave;  // wave owns rows row0 + 8*j

  // pull 118 transformed params for this lane's feature into VGPRs (coalesced per k)
  float prm[P_DIM];
#pragma unroll
  for (int k = 0; k < P_DIM; ++k) prm[k] = pt[k * F_DIM + f];

  const float* g = in + (size_t)row0 * F_DIM + f;

  // kick off async stage of pair 0 (rows j=0 and j=1)
  async_ld32(g,             &stage[wave][0][0][lane]);
  async_ld32(g + 8 * F_DIM, &stage[wave][0][1][lane]);

  for (int i = 0; i < PAIRS; ++i) {
    const int buf = i & 1;
    if (i + 1 < PAIRS) {
      // prefetch next pair while we compute this one; async ops complete in
      // order, so asynccnt<=2 guarantees this iteration's pair has landed.
      const float* gn = g + (size_t)(16 * (i + 1)) * F_DIM;
      async_ld32(gn,             &stage[wave][buf ^ 1][0][lane]);
      async_ld32(gn + 8 * F_DIM, &stage[wave][buf ^ 1][1][lane]);
      asm volatile("s_wait_asynccnt 0x2" ::: "memory");
    } else {
      asm volatile("s_wait_asynccnt 0x0" ::: "memory");
    }

    float xa = stage[wave][buf][0][lane];
    float xb = stage[wave][buf][1][lane];

    // two independent elements in flight -> ILP ~6 across the tanh chains
    float ra = mlp_eval(xa, prm);
    float rb = mlp_eval(xb, prm);

    size_t o = (size_t)(row0 + 16 * i) * F_DIM + f;
    out[o]                     = ra;
    out[o + (size_t)8 * F_DIM] = rb;
  }
}

// ---------------- parameter transform kernel ----------------
// params is (F, P) row-major. Output pt is (P, F): pt[k*F + f] so that the
// main kernel's per-k register fill is a coalesced 128B load per wave.
// k classification (hidden=3, layers=9):
//   k in [0,3)  : layer-0 weights  -> softplus
//   k in [3,6)  : layer-0 bias     -> copy
//   k in [6,9)  : layer-0 gate s   -> tanh
//   k in [9,114): 7 middle layers, 15 each: q=(k-9)%15: q<9 w, q<12 b, else s
//   k in [114,117): final weights  -> softplus
//   k == 117    : final bias       -> copy
__global__ __launch_bounds__(256) void xform_params(const float* __restrict__ p,
                                                    float* __restrict__ pt) {
  int i = blockIdx.x * blockDim.x + threadIdx.x;
  if (i >= P_DIM * F_DIM) return;
  int k = i / F_DIM;
  int f = i - k * F_DIM;
  float v = p[f * P_DIM + k];
  int kind;  // 0 = softplus, 1 = copy, 2 = tanh
  if (k < 9) {
    kind = (k < 3) ? 0 : ((k < 6) ? 1 : 2);
  } else if (k < 114) {
    int q = (k - 9) % 15;
    kind = (q < 9) ? 0 : ((q < 12) ? 1 : 2);
  } else {
    kind = (k < 117) ? 0 : 1;
  }
  float r;
  if (kind == 0) {
    // numerically stable softplus: max(x,0) + log1p(exp(-|x|))
    r = fmaxf(v, 0.0f) + log1pf(expf(-fabsf(v)));
  } else if (kind == 2) {
    r = tanhf(v);
  } else {
    r = v;
  }
  pt[k * F_DIM + f] = r;
}

// ---------------- launch ----------------
extern "C" void kernel_launch(void* const* d_in, const int* in_sizes, int n_in,
                              void* d_out, int out_size, void* d_ws, size_t ws_size,
                              hipStream_t stream) {
  const float* inputs = (const float*)d_in[0];  // (32768, 512) f32
  const float* params = (const float*)d_in[1];  // (512, 118) f32
  float* out = (float*)d_out;                   // (32768, 512) f32
  float* pt  = (float*)d_ws;                    // (118, 512) f32 transformed params

  (void)in_sizes; (void)n_in; (void)out_size; (void)ws_size;

  const int n = P_DIM * F_DIM;
  xform_params<<<(n + 255) / 256, 256, 0, stream>>>(params, pt);

  dim3 grid(F_DIM / 32, B_DIM / ROWS_PER_BLOCK);  // (16, 128)
  cdf_main<<<grid, dim3(256), 0, stream>>>(inputs, pt, out);
}